// GCNLayer_27556510171573
// MI455X (gfx1250) — compile-verified
//
#include <hip/hip_runtime.h>
#include <hip/hip_bf16.h>

typedef float v2f __attribute__((ext_vector_type(2)));
typedef float v8f __attribute__((ext_vector_type(8)));

#define D 64                 // D_IN == D_OUT == 64

// ---------------------------------------------------------------------------
// Kernel 1: zero the accumulator h[N, 64] in workspace (float4 stores).
// ---------------------------------------------------------------------------
__global__ void gcn_zero_f4(float4* __restrict__ p, int n4) {
    int i = blockIdx.x * blockDim.x + threadIdx.x;
    if (i < n4) p[i] = make_float4(0.f, 0.f, 0.f, 0.f);
}

// ---------------------------------------------------------------------------
// Kernel 2: edge scatter.  One wave (32 lanes) per edge; each lane handles a
// contiguous float2 of the 64-float feature row.  src/dst are wave-uniform so
// the index loads scalarize; gather is a coalesced float2; scatter is two
// native GLOBAL_ATOMIC_ADD_F32 ops (inline asm guarantees the hardware f32
// atomic, no CAS loop) that resolve in L2 -- both the feature table (25.6 MB)
// and the accumulator h (25.6 MB) are L2-resident (192 MB).
// ---------------------------------------------------------------------------
__global__ void gcn_edge_scatter(const float* __restrict__ feat,
                                 const int* __restrict__ src,
                                 const int* __restrict__ dst,
                                 float* __restrict__ h,
                                 int nEdges) {
    unsigned gid  = blockIdx.x * blockDim.x + threadIdx.x;
    int      edge = (int)(gid >> 5);
    int      lane = (int)(gid & 31u);
    if (edge >= nEdges) return;

    int s = src[edge];
    int d = dst[edge];

    v2f v = *(const v2f*)(feat + (size_t)s * D + lane * 2);
    float* o = h + (size_t)d * D + lane * 2;

    // Non-returning f32 atomic adds (GLOBAL_ATOMIC_ADD_F32, STOREcnt-tracked).
    asm volatile("global_atomic_add_f32 %0, %1, off\n\t"
                 "global_atomic_add_f32 %0, %2, off offset:4"
                 :
                 : "v"(o), "v"(v.x), "v"(v.y)
                 : "memory");
}

// ---------------------------------------------------------------------------
// Kernel 3: out = h @ W^T + b using V_WMMA_F32_16X16X4_F32.
// One wave computes a 16x64 output slab: 4 accumulators (one per 16-col tile),
// K=64 in steps of 4 -> 16 K-steps, A fragment reused across the 4 WMMAs.
//
// A (16x4 f32, 2 VGPRs/lane as v2f): lane m=lane&15 holds rows, half-wave
//   selects K-pair: a = h[rowBase+m, k0 + 2*(lane>>4) .. +1]   (float2 load)
// B (4x16, B[k][n] = W[n][k]): b = W[ntile*16 + m, k0 + 2*(lane>>4) .. +1]
// D (16x16 f32, 8 VGPRs): col = lane&15, row = v + 8*(lane>>4).
// ---------------------------------------------------------------------------
__global__ void gcn_wmma_gemm_bias(const float* __restrict__ h,
                                   const float* __restrict__ W,
                                   const float* __restrict__ bias,
                                   float* __restrict__ out,
                                   int nRowTiles) {
    int waveId = blockIdx.x * (blockDim.x >> 5) + (threadIdx.x >> 5);
    if (waveId >= nRowTiles) return;   // wave-uniform: whole wave branches

    int lane = threadIdx.x & 31;
    int m    = lane & 15;              // A row within tile / B col (=W row)
    int koff = (lane >> 4) << 1;       // 0 or 2: which K pair this half holds

    size_t rowBase = (size_t)waveId * 16;
    const float* hRow = h + (rowBase + m) * D;
    const float* w0   = W + (size_t)( 0 + m) * D;
    const float* w1   = W + (size_t)(16 + m) * D;
    const float* w2   = W + (size_t)(32 + m) * D;
    const float* w3   = W + (size_t)(48 + m) * D;

    v8f acc0 = {}, acc1 = {}, acc2 = {}, acc3 = {};

#pragma unroll
    for (int k0 = 0; k0 < D; k0 += 4) {
        v2f a  = *(const v2f*)(hRow + k0 + koff);
        v2f b0 = *(const v2f*)(w0   + k0 + koff);
        v2f b1 = *(const v2f*)(w1   + k0 + koff);
        v2f b2 = *(const v2f*)(w2   + k0 + koff);
        v2f b3 = *(const v2f*)(w3   + k0 + koff);
        acc0 = __builtin_amdgcn_wmma_f32_16x16x4_f32(false, a, false, b0,
                                                     (short)0, acc0, false, false);
        acc1 = __builtin_amdgcn_wmma_f32_16x16x4_f32(false, a, false, b1,
                                                     (short)0, acc1, false, false);
        acc2 = __builtin_amdgcn_wmma_f32_16x16x4_f32(false, a, false, b2,
                                                     (short)0, acc2, false, false);
        acc3 = __builtin_amdgcn_wmma_f32_16x16x4_f32(false, a, false, b3,
                                                     (short)0, acc3, false, false);
    }

    float bv0 = bias[ 0 + m];
    float bv1 = bias[16 + m];
    float bv2 = bias[32 + m];
    float bv3 = bias[48 + m];

    int rh = (lane >> 4) * 8;          // row offset of this half-wave's VGPRs
#pragma unroll
    for (int v = 0; v < 8; ++v) {
        size_t row = rowBase + rh + v;
        float* orow = out + row * D;
        orow[ 0 + m] = acc0[v] + bv0;
        orow[16 + m] = acc1[v] + bv1;
        orow[32 + m] = acc2[v] + bv2;
        orow[48 + m] = acc3[v] + bv3;
    }
}

// ---------------------------------------------------------------------------
// Launcher
// ---------------------------------------------------------------------------
extern "C" void kernel_launch(void* const* d_in, const int* in_sizes, int n_in,
                              void* d_out, int out_size, void* d_ws, size_t ws_size,
                              hipStream_t stream) {
    const float* feature = (const float*)d_in[0];   // [N, 64] f32
    const int*   src     = (const int*)  d_in[1];   // [E] i32
    const int*   dst     = (const int*)  d_in[2];   // [E] i32
    const float* W       = (const float*)d_in[3];   // [64, 64] f32
    const float* bias    = (const float*)d_in[4];   // [64] f32
    float*       out     = (float*)d_out;           // [N, 64] f32
    float*       h       = (float*)d_ws;            // scratch accumulator [N, 64]

    int nNodes = in_sizes[0] / D;                   // 100000
    int nEdges = in_sizes[1];                       // 1600000

    // 1) h = 0
    int n4 = (nNodes * D) / 4;
    gcn_zero_f4<<<(n4 + 255) / 256, 256, 0, stream>>>((float4*)h, n4);

    // 2) h[dst] += feature[src]   (one wave per edge, 2 floats per lane)
    long long totalThreads = (long long)nEdges * 32;
    int blocks = (int)((totalThreads + 255) / 256);
    gcn_edge_scatter<<<blocks, 256, 0, stream>>>(feature, src, dst, h, nEdges);

    // 3) out = h @ W^T + b   (WMMA f32 16x16x4, one wave per 16x64 slab)
    int rowTiles = (nNodes + 15) / 16;              // 6250 (exact)
    int gemmBlocks = (rowTiles + 7) / 8;            // 8 waves per 256-thread block
    gcn_wmma_gemm_bias<<<gemmBlocks, 256, 0, stream>>>(h, W, bias, out, rowTiles);
}